// ScaffoldLoss_79001628443369
// MI455X (gfx1250) — compile-verified
//
#include <hip/hip_runtime.h>
#include <math.h>
#include <stdint.h>

// Problem constants (match reference)
#define B_   4
#define N_   2048
#define M_   2048
#define M2_  4096
#define D_   128
#define K_   512

typedef float v2f __attribute__((ext_vector_type(2)));
typedef float v8f __attribute__((ext_vector_type(8)));

__constant__ float LIGQ_C[10]  = {0.0f,-0.3f,-0.4f,-0.1f,0.1f,-0.2f,-0.1f,-0.1f,-0.1f,0.1f};
__constant__ float PROTQ_C[4]  = {0.0f,-0.3f,-0.4f,-0.2f};

// ---- workspace layout (floats) ----
#define OFF_LIGQ   0          // 8192
#define OFF_PROTQ  8192       // 16384
#define OFF_ESQ    24576      // 512
#define OFF_ROW    25088      // 8192
#define OFF_COL    33280      // 8192
#define OFF_VQ     41472      // 512
#define OFF_YY     41984      // 16384
#define OFF_CR     58368      // 2*32768

// LDS E-tile row stride (floats): 132 = 128 + 4 pad.
// 4*lx bank offset per row-group lane -> conflict-free; 528B row pitch keeps
// every b128 async destination 16B-aligned.
#define ET_STRIDE 132

// Precompute charges (argmax of leading feature channels) and |e_k|^2.
__global__ void prep_kernel(const float* __restrict__ sf, const float* __restrict__ pf,
                            const float* __restrict__ emb, float* __restrict__ ligq,
                            float* __restrict__ protq, float* __restrict__ esq) {
    int gid = blockIdx.x * blockDim.x + threadIdx.x;
    if (gid < B_ * M_) {
        const float* f = sf + (size_t)gid * D_;
        int bi = 0; float bv = f[0];
        #pragma unroll
        for (int t = 1; t < 10; ++t) { float v = f[t]; if (v > bv) { bv = v; bi = t; } }
        ligq[gid] = LIGQ_C[bi];
    } else if (gid < B_ * M_ + B_ * M2_) {
        int i = gid - B_ * M_;
        const float* f = pf + (size_t)i * D_;
        int bi = 0; float bv = f[0];
        #pragma unroll
        for (int t = 1; t < 4; ++t) { float v = f[t]; if (v > bv) { bv = v; bi = t; } }
        protq[i] = PROTQ_C[bi];
    } else if (gid < B_ * M_ + B_ * M2_ + K_) {
        int k = gid - (B_ * M_ + B_ * M2_);
        const float* e = emb + (size_t)k * D_;
        float s = 0.f;
        for (int d = 0; d < D_; ++d) { float v = e[d]; s += v * v; }
        esq[k] = s;
    }
}

// Per-(b,i) min distance from P[b][i] to all Q[b][*].
// Q tiles are double-buffered through LDS via GLOBAL_LOAD_ASYNC_TO_LDS_B32
// (ASYNCcnt pipeline): issue next tile, compute current one.
__global__ void min_dist_kernel(const float* __restrict__ P, const float* __restrict__ Q,
                                float* __restrict__ outmin, int nP, int nQ) {
    __shared__ float qs[2][256 * 3];
    int tid = threadIdx.x;
    int gid = blockIdx.x * 256 + tid;
    int b = gid / nP;                       // block covers contiguous i within one b
    float px = P[(size_t)gid * 3 + 0];
    float py = P[(size_t)gid * 3 + 1];
    float pz = P[(size_t)gid * 3 + 2];
    const float* qbase = Q + (size_t)b * nQ * 3;

    // prologue: async-copy tile 0 into buffer 0 (3 coalesced b32 per thread)
    #pragma unroll
    for (int k = 0; k < 3; ++k) {
        unsigned lds = (unsigned)(uintptr_t)(&qs[0][k * 256 + tid]);
        const float* gp = qbase + (k * 256 + tid);
        asm volatile("global_load_async_to_lds_b32 %0, %1, off"
                     :: "v"(lds), "v"(gp) : "memory");
    }

    float best = 3.4e38f;
    int ntiles = nQ / 256;
    for (int t = 0; t < ntiles; ++t) {
        asm volatile("s_wait_asynccnt 0x0" ::: "memory");
        __syncthreads();                    // tile t visible block-wide
        if (t + 1 < ntiles) {               // issue tile t+1 into other buffer
            #pragma unroll
            for (int k = 0; k < 3; ++k) {
                unsigned lds = (unsigned)(uintptr_t)(&qs[(t + 1) & 1][k * 256 + tid]);
                const float* gp = qbase + ((size_t)(t + 1) * 768 + k * 256 + tid);
                asm volatile("global_load_async_to_lds_b32 %0, %1, off"
                             :: "v"(lds), "v"(gp) : "memory");
            }
        }
        const float* qt = qs[t & 1];
        #pragma unroll 8
        for (int j = 0; j < 256; ++j) {
            float dx = px - qt[j * 3 + 0];
            float dy = py - qt[j * 3 + 1];
            float dz = pz - qt[j * 3 + 2];
            float d2 = dx * dx + dy * dy + dz * dz;
            best = fminf(best, d2);
        }
        // no trailing barrier needed: top-of-loop wait+barrier orders everything
    }
    outmin[gid] = sqrtf(best);
}

// VQ: G = F(8192x128) @ E^T(128x512) via V_WMMA_F32_16X16X4_F32, per-row argmin of
// (|e|^2 - 2G), then sum of (e_sel - f)^2 over the row.
// Block = 8 waves x 16 rows = 128 rows. A-fragments hoisted to registers once.
// E column tiles (8KB each) are async-copied (B128) into a double-buffered,
// bank-conflict-free LDS tile shared by all 8 waves; B-fragments come from LDS.
__global__ void __launch_bounds__(256) vq_kernel(const float* __restrict__ F,
                                                 const float* __restrict__ E,
                                                 const float* __restrict__ esq,
                                                 float* __restrict__ part) {
    __shared__ float et[2][16 * ET_STRIDE];
    int tid  = threadIdx.x;
    int lane = tid & 31;
    int wave = tid >> 5;
    int w = blockIdx.x * 8 + wave;       // 512 waves total
    int rowbase = w * 16;
    int hi = lane >> 4;                  // half-wave select
    int lx = lane & 15;

    // A layout (16x4 f32): lanes 0-15 hold K=0/1, lanes 16-31 hold K=2/3
    const float* frow = F + (size_t)(rowbase + lx) * D_ + 2 * hi;
    float a0[32], a1[32];
    #pragma unroll
    for (int dc = 0; dc < 32; ++dc) { a0[dc] = frow[4 * dc]; a1[dc] = frow[4 * dc + 1]; }

    // async E-tile copy: thread t moves 8 floats (2 x b128) of the 16x128 tile
    int erow_t = tid >> 4;               // 0..15  (tile row)
    int ecol_t = (tid & 15) * 8;         // 0..120 (tile col base)
    #pragma unroll
    for (int k = 0; k < 2; ++k) {
        unsigned lds = (unsigned)(uintptr_t)(&et[0][erow_t * ET_STRIDE + ecol_t + k * 4]);
        const float* gp = E + (size_t)erow_t * D_ + ecol_t + k * 4;   // tile 0
        asm volatile("global_load_async_to_lds_b128 %0, %1, off"
                     :: "v"(lds), "v"(gp) : "memory");
    }

    float best_val[8];
    int   best_col[8];
    #pragma unroll
    for (int v = 0; v < 8; ++v) { best_val[v] = 3.4e38f; best_col[v] = 0; }

    for (int ct = 0; ct < K_ / 16; ++ct) {
        asm volatile("s_wait_asynccnt 0x0" ::: "memory");
        __syncthreads();                 // E tile ct visible block-wide
        if (ct + 1 < K_ / 16) {          // prefetch tile ct+1 into other buffer
            #pragma unroll
            for (int k = 0; k < 2; ++k) {
                unsigned lds = (unsigned)(uintptr_t)(
                    &et[(ct + 1) & 1][erow_t * ET_STRIDE + ecol_t + k * 4]);
                const float* gp = E + (size_t)(ct + 1) * 16 * D_
                                    + (size_t)erow_t * D_ + ecol_t + k * 4;
                asm volatile("global_load_async_to_lds_b128 %0, %1, off"
                             :: "v"(lds), "v"(gp) : "memory");
            }
        }

        const float* erow = &et[ct & 1][lx * ET_STRIDE + 2 * hi];  // B frag from LDS
        v8f c = {0.f, 0.f, 0.f, 0.f, 0.f, 0.f, 0.f, 0.f};
        #pragma unroll
        for (int dc = 0; dc < 32; ++dc) {
            v2f a, bb;
            a.x  = a0[dc];          a.y  = a1[dc];
            bb.x = erow[4 * dc];    bb.y = erow[4 * dc + 1];
            c = __builtin_amdgcn_wmma_f32_16x16x4_f32(false, a, false, bb,
                                                      (short)0, c, false, false);
        }
        int col = ct * 16 + lx;
        float ec = esq[col];
        #pragma unroll
        for (int v = 0; v < 8; ++v) {
            // c[v] = G[rowbase + v + 8*hi][col]
            float val = ec - 2.0f * c[v];
            int bc = col;
            #pragma unroll
            for (int off = 1; off < 16; off <<= 1) {   // reduce within 16-lane row group
                float ov = __shfl_xor(val, off, 32);
                int   oc = __shfl_xor(bc,  off, 32);
                bool take = (ov < val) || ((ov == val) && (oc < bc));
                val = take ? ov : val;                 // branchless -> v_cndmask
                bc  = take ? oc : bc;
            }
            bool tk = (val < best_val[v]);             // strict < keeps earliest tile
            best_val[v] = tk ? val : best_val[v];
            best_col[v] = tk ? bc  : best_col[v];
        }
        // WAR on buffer (ct+1)&1 is safe: all threads passed this iteration's
        // barrier only after finishing compute of ct-1 (its last reader).
    }

    // (1+beta)*mean handled in final kernel; here: raw sum of squared diffs.
    float acc = 0.f;
    #pragma unroll
    for (int v = 0; v < 8; ++v) {
        int row = rowbase + v + 8 * hi;
        int idx = best_col[v];
        const float* fr = F + (size_t)row * D_;
        const float* er = E + (size_t)idx * D_;
        for (int d = lx; d < D_; d += 16) {
            float df = er[d] - fr[d];
            acc += df * df;
        }
    }
    #pragma unroll
    for (int off = 1; off < 32; off <<= 1) acc += __shfl_xor(acc, off, 32);
    if (lane == 0) part[w] = acc;
}

// sum over b,i,j of (|y_i - y_j| - Ts[i,j])^2 ; template read exactly once.
__global__ void yy_kernel(const float* __restrict__ y, const float* __restrict__ Ts,
                          float* __restrict__ part) {
    int tid = threadIdx.x;
    int jb = blockIdx.x % (M_ / 16);
    int ib = blockIdx.x / (M_ / 16);
    int i = ib * 16 + (tid >> 4);
    int j = jb * 16 + (tid & 15);
    float T = Ts[(size_t)i * M_ + j];
    float acc = 0.f;
    #pragma unroll
    for (int b = 0; b < B_; ++b) {
        size_t yi = ((size_t)b * M_ + i) * 3;
        size_t yj = ((size_t)b * M_ + j) * 3;
        float dx = y[yi + 0] - y[yj + 0];
        float dy = y[yi + 1] - y[yj + 1];
        float dz = y[yi + 2] - y[yj + 2];
        float d = sqrtf(dx * dx + dy * dy + dz * dz);
        float e = d - T;
        acc += e * e;
    }
    __shared__ float red[256];
    red[tid] = acc; __syncthreads();
    for (int s = 128; s > 0; s >>= 1) { if (tid < s) red[tid] += red[tid + s]; __syncthreads(); }
    if (tid == 0) part[blockIdx.x] = red[0];
}

// Fused: (d_yp - Tc)^2  AND  elec/vdw on dd = d_yp + delta. One pass over d_yp.
__global__ void cross_kernel(const float* __restrict__ y, const float* __restrict__ p,
                             const float* __restrict__ Tc, const float* __restrict__ ligq,
                             const float* __restrict__ protq, float* __restrict__ part) {
    int tid = threadIdx.x;
    int jb = blockIdx.x % (M2_ / 16);
    int ib = blockIdx.x / (M2_ / 16);
    int i = ib * 16 + (tid >> 4);
    int j = jb * 16 + (tid & 15);
    float T = Tc[(size_t)i * M2_ + j];
    float acc_t = 0.f, acc_e = 0.f;
    #pragma unroll
    for (int b = 0; b < B_; ++b) {
        size_t yi = ((size_t)b * M_  + i) * 3;
        size_t pj = ((size_t)b * M2_ + j) * 3;
        float dx = y[yi + 0] - p[pj + 0];
        float dy = y[yi + 1] - p[pj + 1];
        float dz = y[yi + 2] - p[pj + 2];
        float d = sqrtf(dx * dx + dy * dy + dz * dz);
        float e = d - T; acc_t += e * e;
        float dd = d + 0.01f;
        float inv = 1.0f / dd;
        acc_e += ligq[b * M_ + i] * protq[b * M2_ + j] * inv;   // EPSILON = 1
        float inv2 = inv * inv;
        float inv6 = inv2 * inv2 * inv2;
        acc_e += inv6 * inv6 - inv6;                            // A_VDW = B_VDW = 1
    }
    __shared__ float red[512];
    red[tid] = acc_t; red[256 + tid] = acc_e; __syncthreads();
    for (int s = 128; s > 0; s >>= 1) {
        if (tid < s) { red[tid] += red[tid + s]; red[256 + tid] += red[256 + tid + s]; }
        __syncthreads();
    }
    if (tid == 0) { part[2 * blockIdx.x] = red[0]; part[2 * blockIdx.x + 1] = red[256]; }
}

// Deterministic single-block combine.
__global__ void final_kernel(const float* __restrict__ ws, float* __restrict__ out) {
    int tid = threadIdx.x;
    float sRow = 0, sCol = 0, sVq = 0, sYy = 0, sCt = 0, sEv = 0;
    for (int k = tid; k < 8192;  k += 256) sRow += ws[OFF_ROW + k];
    for (int k = tid; k < 8192;  k += 256) sCol += ws[OFF_COL + k];
    for (int k = tid; k < 512;   k += 256) sVq  += ws[OFF_VQ  + k];
    for (int k = tid; k < 16384; k += 256) sYy  += ws[OFF_YY  + k];
    for (int k = tid; k < 32768; k += 256) { sCt += ws[OFF_CR + 2 * k]; sEv += ws[OFF_CR + 2 * k + 1]; }

    __shared__ float red[256];
    auto reduce = [&](float v) -> float {
        red[tid] = v; __syncthreads();
        for (int s = 128; s > 0; s >>= 1) { if (tid < s) red[tid] += red[tid + s]; __syncthreads(); }
        float r = red[0]; __syncthreads();
        return r;
    };
    float Srow = reduce(sRow), Scol = reduce(sCol), Svq = reduce(sVq);
    float Syy  = reduce(sYy),  Sct  = reduce(sCt),  Sev = reduce(sEv);

    if (tid == 0) {
        float chamfer = (Srow + Scol) / (float)(B_ * N_);
        float vq      = 1.25f * Svq / (float)(B_ * M_ * D_);           // (1+beta)*mean
        float tpl     = Syy / (float)((double)B_ * M_ * M_)
                      + Sct / (float)((double)B_ * N_ * M2_);
        float elec    = 0.1f * (Sev / (float)B_);                      // LAMBDA_ELEC * mean_b
        out[0] = chamfer + vq + tpl + elec;
    }
}

extern "C" void kernel_launch(void* const* d_in, const int* in_sizes, int n_in,
                              void* d_out, int out_size, void* d_ws, size_t ws_size,
                              hipStream_t stream) {
    (void)in_sizes; (void)n_in; (void)out_size; (void)ws_size;
    const float* x   = (const float*)d_in[0];
    const float* y   = (const float*)d_in[1];
    const float* enc = (const float*)d_in[2];
    const float* pc  = (const float*)d_in[3];
    const float* sf  = (const float*)d_in[4];
    const float* pf  = (const float*)d_in[5];
    const float* Ts  = (const float*)d_in[6];
    const float* Tc  = (const float*)d_in[7];
    const float* emb = (const float*)d_in[8];
    float* ws  = (float*)d_ws;
    float* out = (float*)d_out;

    prep_kernel<<<98, 256, 0, stream>>>(sf, pf, emb,
                                        ws + OFF_LIGQ, ws + OFF_PROTQ, ws + OFF_ESQ);
    min_dist_kernel<<<(B_ * N_) / 256, 256, 0, stream>>>(x, y, ws + OFF_ROW, N_, M_);
    min_dist_kernel<<<(B_ * M_) / 256, 256, 0, stream>>>(y, x, ws + OFF_COL, M_, N_);
    vq_kernel<<<64, 256, 0, stream>>>(enc, emb, ws + OFF_ESQ, ws + OFF_VQ);
    yy_kernel<<<(M_ / 16) * (M_ / 16), 256, 0, stream>>>(y, Ts, ws + OFF_YY);
    cross_kernel<<<(N_ / 16) * (M2_ / 16), 256, 0, stream>>>(y, pc, Tc,
                                                             ws + OFF_LIGQ, ws + OFF_PROTQ,
                                                             ws + OFF_CR);
    final_kernel<<<1, 256, 0, stream>>>(ws, out);
}